// GraphFeature_5257039970655
// MI455X (gfx1250) — compile-verified
//
#include <hip/hip_runtime.h>

typedef float v2f  __attribute__((ext_vector_type(2)));
typedef float v8f  __attribute__((ext_vector_type(8)));
typedef float vf4  __attribute__((ext_vector_type(4)));

#define C        64
#define KNN      20
#define NPTS     4096
#define TILE     16
#define WPB      4               // waves per block (all same batch: 256 % 4 == 0)
#define BROW     68              // padded LDS row stride for the shared B tile
#define TROW     17              // padded row stride for the distance tile
#define NEG_INF  (-3.402823466e38f)

__global__ __launch_bounds__(WPB * 32)
void knn_edge_kernel(const float* __restrict__ x, float* __restrict__ out) {
    __shared__ float lds_B[TILE * BROW];             // block-shared 16x64 column tile
    __shared__ float lds_tile[WPB][TILE * TROW];     // per-wave 16x16 distance tile
    __shared__ float lds_vals[WPB][32][KNN];         // per-lane sorted top-K values
    __shared__ int   lds_idxs[WPB][32][KNN];         // per-lane top-K indices
    __shared__ float lds_sq  [WPB][TILE];            // row norms of the A tile
    __shared__ int   lds_sel [WPB][TILE * KNN];      // merged neighbor indices

    const int tid  = threadIdx.x;
    const int lane = tid & 31;
    const int wave = tid >> 5;
    const int tile_id = blockIdx.x * WPB + wave;
    const int tiles_per_batch = NPTS / TILE;         // 256
    const int b  = tile_id / tiles_per_batch;        // uniform across the block
    const int i0 = (tile_id % tiles_per_batch) * TILE;

    const float* __restrict__ xb = x + (size_t)b * NPTS * C;

    const int m    = lane & 15;   // M index (A) / N index (B) within tile
    const int half = lane >> 4;   // selects K pair within a 4-wide chunk

    // ---- Load A fragments (batched: 16 independent b64 loads, one wait) ----
    v2f a[16];
    {
        const float* rp = xb + (size_t)(i0 + m) * C;
        #pragma unroll
        for (int k = 0; k < 16; ++k)
            a[k] = *(const v2f*)(rp + 4 * k + 2 * half);
    }
    float sq_half = 0.f;
    #pragma unroll
    for (int k = 0; k < 16; ++k)
        sq_half += a[k].x * a[k].x + a[k].y * a[k].y;
    float sq_row_own = sq_half + __shfl_xor(sq_half, 16, 32);
    if (lane < TILE) lds_sq[wave][m] = sq_row_own;
    float srow[8];
    #pragma unroll
    for (int r = 0; r < 8; ++r) srow[r] = lds_sq[wave][r + 8 * half];

    // ---- Init per-lane top-K lists (sorted descending) ----
    #pragma unroll
    for (int t = 0; t < KNN; ++t) {
        lds_vals[wave][lane][t] = NEG_INF;
        lds_idxs[wave][lane][t] = 0;
    }
    float curMin = NEG_INF;

    // ---- Software-pipelined staging of the shared B tile ----
    const int f0 = 4 * tid;            // floats [f0..f0+3] and [f0+512..]
    const int r0 = f0 >> 6, c0 = f0 & 63;
    const int r1 = (f0 + 512) >> 6;    // same c0
    vf4 st0 = *(const vf4*)(xb + f0);
    vf4 st1 = *(const vf4*)(xb + f0 + 512);

    for (int j0 = 0; j0 < NPTS; j0 += TILE) {
        __syncthreads();                               // prev tile fully consumed
        *(vf4*)&lds_B[r0 * BROW + c0] = st0;
        *(vf4*)&lds_B[r1 * BROW + c0] = st1;
        __syncthreads();                               // tile visible to all waves

        if (j0 + TILE < NPTS) {                        // prefetch next tile -> regs
            const float* np = xb + (size_t)(j0 + TILE) * C;
            st0 = *(const vf4*)(np + f0);
            st1 = *(const vf4*)(np + f0 + 512);
        }

        // Batched conflict-free fragment reads from LDS, then chained WMMAs
        v2f bf[16];
        #pragma unroll
        for (int k = 0; k < 16; ++k)
            bf[k] = *(const v2f*)&lds_B[m * BROW + 4 * k + 2 * half];

        v8f acc = {0.f, 0.f, 0.f, 0.f, 0.f, 0.f, 0.f, 0.f};
        float bsq = 0.f;
        #pragma unroll
        for (int k = 0; k < 16; ++k) {
            bsq += bf[k].x * bf[k].x + bf[k].y * bf[k].y;
            acc = __builtin_amdgcn_wmma_f32_16x16x4_f32(
                      false, a[k], false, bf[k], (short)0, acc, false, false);
        }
        float scol = bsq + __shfl_xor(bsq, 16, 32);    // |col|^2

        // C/D layout: VGPR r, lane l -> row = r + 8*(l/16), col = l%16
        #pragma unroll
        for (int r = 0; r < 8; ++r) {
            int row_local = r + 8 * half;
            lds_tile[wave][row_local * TROW + m] = 2.f * acc[r] - srow[r] - scol;
        }

        // Each lane scans 8 columns of its row (lane%16); halves split by lane/16
        const int colbase = 8 * half;
        #pragma unroll
        for (int i = 0; i < 8; ++i) {
            float v   = lds_tile[wave][m * TROW + colbase + i];
            int   idx = j0 + colbase + i;
            if (v > curMin) {
                int t = KNN - 1;
                while (t > 0 && lds_vals[wave][lane][t - 1] < v) {
                    lds_vals[wave][lane][t] = lds_vals[wave][lane][t - 1];
                    lds_idxs[wave][lane][t] = lds_idxs[wave][lane][t - 1];
                    --t;
                }
                lds_vals[wave][lane][t] = v;
                lds_idxs[wave][lane][t] = idx;
                curMin = lds_vals[wave][lane][KNN - 1];
            }
        }
    }

    // ---- Merge the two half-row lists (both sorted desc); tie -> lower index ----
    if (lane < TILE) {
        const float* va = lds_vals[wave][lane];
        const int*   ia = lds_idxs[wave][lane];
        const float* vb = lds_vals[wave][lane + 16];
        const int*   ib = lds_idxs[wave][lane + 16];
        int pa = 0, pb = 0;
        for (int t = 0; t < KNN; ++t) {
            float av = va[pa], bv = vb[pb];
            bool takeA = (av > bv) || (av == bv && ia[pa] < ib[pb]);
            lds_sel[wave][lane * KNN + t] = takeA ? ia[pa] : ib[pb];
            if (takeA) ++pa; else ++pb;
        }
    }

    // ---- Cooperative gather + write: out[b,n,k,0:64]=x[nb], [64:128]=x[n] ----
    const int cw = (lane & 15) * 4;
    for (int p = 0; p < TILE * KNN; ++p) {
        int row = p / KNN;
        int kk  = p - row * KNN;
        int nb  = lds_sel[wave][p];
        int src = half ? (i0 + row) : nb;    // lanes 0-15: neighbor, 16-31: self
        vf4 v = *(const vf4*)(xb + (size_t)src * C + cw);
        size_t o = (((size_t)b * NPTS + (i0 + row)) * KNN + kk) * (size_t)(2 * C)
                   + (size_t)half * C + cw;
        *(vf4*)(out + o) = v;
    }
}

extern "C" void kernel_launch(void* const* d_in, const int* in_sizes, int n_in,
                              void* d_out, int out_size, void* d_ws, size_t ws_size,
                              hipStream_t stream) {
    const float* x = (const float*)d_in[0];
    float* out = (float*)d_out;
    int total_rows = in_sizes[0] / C;            // B * N = 32768
    int tiles  = total_rows / TILE;              // 2048 row tiles (one wave each)
    int blocks = tiles / WPB;                    // 512 blocks of 4 waves
    knn_edge_kernel<<<blocks, WPB * 32, 0, stream>>>(x, out);
}